// GraphSAGE_11141145166397
// MI455X (gfx1250) — compile-verified
//
#include <hip/hip_runtime.h>
#include <hip/hip_bf16.h>

#define N_NODES 100000
#define N_EDGES 1600000
#define D_IN    64
#define D_HID   128
#define D_OUT   64
#define BN_EPS  1e-5f

typedef float v2f __attribute__((ext_vector_type(2)));
typedef float v8f __attribute__((ext_vector_type(8)));

// ---------------------------------------------------------------------------
// Scatter: one thread per (edge, 4-feature chunk). Gathers a float4 from the
// source node's row (L2-resident) and atomically accumulates into agg[dst].
// Chunk 0 also bumps the degree counter (layer 1 only).
// ---------------------------------------------------------------------------
template<int SHIFT, bool ADD_DEG>
__global__ __launch_bounds__(256)
void sage_scatter_kernel(const float* __restrict__ feat,
                         const int*   __restrict__ src,
                         const int*   __restrict__ dst,
                         float* __restrict__ agg,
                         float* __restrict__ deg,
                         int nedges, int dim)
{
    const long tid    = (long)blockIdx.x * blockDim.x + threadIdx.x;
    const long total  = (long)nedges << SHIFT;
    if (tid >= total) return;
    const int e = (int)(tid >> SHIFT);
    const int c = (int)(tid & ((1 << SHIFT) - 1));
    const int s = src[e];
    const int d = dst[e];
    const float4 v = *(const float4*)(feat + (long)s * dim + (c << 2));
    float* a = agg + (long)d * dim + (c << 2);
    atomicAdd(a + 0, v.x);
    atomicAdd(a + 1, v.y);
    atomicAdd(a + 2, v.z);
    atomicAdd(a + 3, v.w);
    if (ADD_DEG && c == 0) atomicAdd(deg + d, 1.0f);
}

// ---------------------------------------------------------------------------
// Fused node kernel: out = BN( [agg/deg | x] @ [Wl;Wr]^T + bl ) (+ReLU)
//   - A tile ([16*RG] x KCAT, KCAT = 2*DIN) staged in LDS with mean division
//   - B = W_cat^T (KCAT x DOUT) staged in LDS
//   - each wave computes one 16x16 output tile with V_WMMA_F32_16X16X4_F32
// Block = RG*CT waves (RG row-groups of 16 nodes, CT col-tiles of 16 outputs).
// Grid divides N exactly -> no divergence, EXEC all-ones for WMMA.
// ---------------------------------------------------------------------------
template<int DIN, int DOUTT, int RG, int CT, bool RELU>
__global__ __launch_bounds__(RG*CT*32)
void sage_node_kernel(const float* __restrict__ xin,   // [N, DIN]
                      const float* __restrict__ agg,   // [N, DIN] (sums)
                      const float* __restrict__ deg,   // [N]
                      const float* __restrict__ wl,    // [DOUTT, DIN]
                      const float* __restrict__ wr,    // [DOUTT, DIN]
                      const float* __restrict__ bl,    // [DOUTT]
                      const float* __restrict__ bn_g,
                      const float* __restrict__ bn_b,
                      const float* __restrict__ bn_m,
                      const float* __restrict__ bn_v,
                      float* __restrict__ out)         // [N, DOUTT]
{
    constexpr int KCAT    = 2 * DIN;
    constexpr int MROWS   = 16 * RG;
    constexpr int ASTRIDE = KCAT + 2;          // pad to spread LDS banks
    constexpr int NTHR    = RG * CT * 32;

    extern __shared__ float smem[];
    float* As = smem;                          // [MROWS][ASTRIDE]
    float* Bs = smem + MROWS * ASTRIDE;        // [KCAT][DOUTT]

    const int tid   = threadIdx.x;
    const int node0 = blockIdx.x * MROWS;

    // ---- stage B = W_cat^T into LDS (coalesced LDS writes; W reads strided by DIN) ----
    for (int i = tid; i < KCAT * DOUTT; i += NTHR) {
        const int k = i / DOUTT;
        const int o = i % DOUTT;
        const float w = (k < DIN) ? wl[o * DIN + k] : wr[o * DIN + (k - DIN)];
        Bs[k * DOUTT + o] = w;
    }
    // ---- stage A = [agg*inv_deg | x] into LDS ----
    for (int i = tid; i < MROWS * KCAT; i += NTHR) {
        const int r = i / KCAT;
        const int k = i % KCAT;
        const long node = node0 + r;
        float v;
        if (k < DIN) {
            const float inv = 1.0f / fmaxf(deg[node], 1.0f);
            v = agg[node * DIN + k] * inv;
        } else {
            v = xin[node * DIN + (k - DIN)];
        }
        As[r * ASTRIDE + k] = v;
    }
    __syncthreads();

    const int wave  = tid >> 5;
    const int lane  = tid & 31;
    const int rg    = wave / CT;
    const int ct    = wave % CT;
    const int n0    = ct * 16;
    const int mn    = lane & 15;          // M index for A frag, N index for B frag
    const int khalf = (lane >> 4) * 2;    // lanes 16-31 hold K+2, K+3

    v8f c = {};
    #pragma unroll 4
    for (int k0 = 0; k0 < KCAT; k0 += 4) {
        v2f a, b;
        const float* ap = &As[(rg * 16 + mn) * ASTRIDE + k0 + khalf];
        a.x = ap[0];
        a.y = ap[1];
        const float* bp = &Bs[(k0 + khalf) * DOUTT + n0 + mn];
        b.x = bp[0];
        b.y = bp[DOUTT];
        // D = A(16x4,f32) x B(4x16,f32) + C  -> v_wmma_f32_16x16x4_f32
        c = __builtin_amdgcn_wmma_f32_16x16x4_f32(
                /*neg_a=*/false, a, /*neg_b=*/false, b,
                /*c_mod=*/(short)0, c, /*reuse_a=*/false, /*reuse_b=*/false);
    }

    // ---- epilogue: bias + BN (+ReLU), D layout: vgpr g -> M=g (+8 for hi lanes)
    const int o     = n0 + mn;
    const float bias = bl[o];
    const float sc   = __frsqrt_rn(bn_v[o] + BN_EPS) * bn_g[o];
    const float mu   = bn_m[o];
    const float bt   = bn_b[o];
    const int rowbase = node0 + rg * 16 + ((lane >> 4) << 3);
    #pragma unroll
    for (int g = 0; g < 8; ++g) {
        float v = c[g] + bias;
        v = (v - mu) * sc + bt;
        if (RELU) v = fmaxf(v, 0.0f);
        out[(long)(rowbase + g) * DOUTT + o] = v;
    }
}

// ---------------------------------------------------------------------------
extern "C" void kernel_launch(void* const* d_in, const int* in_sizes, int n_in,
                              void* d_out, int out_size, void* d_ws, size_t ws_size,
                              hipStream_t stream) {
    const float* x    = (const float*)d_in[0];
    const int*   ei   = (const int*)d_in[1];
    const float* w1l  = (const float*)d_in[2];
    const float* b1l  = (const float*)d_in[3];
    const float* w1r  = (const float*)d_in[4];
    const float* g1   = (const float*)d_in[5];
    const float* be1  = (const float*)d_in[6];
    const float* m1   = (const float*)d_in[7];
    const float* v1   = (const float*)d_in[8];
    const float* w2l  = (const float*)d_in[9];
    const float* b2l  = (const float*)d_in[10];
    const float* w2r  = (const float*)d_in[11];
    const float* g2   = (const float*)d_in[12];
    const float* be2  = (const float*)d_in[13];
    const float* m2   = (const float*)d_in[14];
    const float* v2   = (const float*)d_in[15];

    const int* src = ei;            // edge_index[0]
    const int* dst = ei + N_EDGES;  // edge_index[1]

    // workspace layout (floats): deg | agg1 | agg2 | h1
    float* ws   = (float*)d_ws;
    float* deg  = ws;
    float* agg1 = deg  + N_NODES;
    float* agg2 = agg1 + (size_t)N_NODES * D_IN;
    float* h1   = agg2 + (size_t)N_NODES * D_HID;
    float* out  = (float*)d_out;

    // zero deg+agg1+agg2 (contiguous) every call (deterministic, capture-safe)
    hipMemsetAsync(deg, 0, sizeof(float) * (size_t)N_NODES * (1 + D_IN + D_HID),
                   stream);

    // ---- layer 1 scatter (+degree): E * (64/4) threads ----
    {
        const long total = (long)N_EDGES * (D_IN / 4);
        const int blocks = (int)((total + 255) / 256);
        sage_scatter_kernel<4, true><<<blocks, 256, 0, stream>>>(
            x, src, dst, agg1, deg, N_EDGES, D_IN);
    }
    // ---- layer 1 node GEMM: 16 nodes x 128 outs per block, 8 waves ----
    {
        const size_t smem = (16 * (2 * D_IN + 2) + (2 * D_IN) * D_HID) * sizeof(float);
        sage_node_kernel<D_IN, D_HID, 1, 8, true>
            <<<N_NODES / 16, 256, smem, stream>>>(
                x, agg1, deg, w1l, w1r, b1l, g1, be1, m1, v1, h1);
    }
    // ---- layer 2 scatter: E * (128/4) threads ----
    {
        const long total = (long)N_EDGES * (D_HID / 4);
        const int blocks = (int)((total + 255) / 256);
        sage_scatter_kernel<5, false><<<blocks, 256, 0, stream>>>(
            h1, src, dst, agg2, nullptr, N_EDGES, D_HID);
    }
    // ---- layer 2 node GEMM: 32 nodes x 64 outs per block, 8 waves ----
    {
        const size_t smem = (32 * (2 * D_HID + 2) + (2 * D_HID) * D_OUT) * sizeof(float);
        sage_node_kernel<D_HID, D_OUT, 2, 4, false>
            <<<N_NODES / 32, 256, smem, stream>>>(
                h1, agg2, deg, w2l, w2r, b2l, g2, be2, m2, v2, out);
    }
}